// RSLP_LIF_45973329937063
// MI455X (gfx1250) — compile-verified
//
#include <hip/hip_runtime.h>

// Problem constants (match reference)
#define BB    64
#define TT    128
#define NN    2048
#define DIN   256
#define DINIT 256
#define DOUT  512
#define DECAY 0.1f

#define KCH   64    // K-chunk staged through LDS
#define PADB  68    // padded k-stride (words): bank = (4n + c) % 64 -> conflict-free

typedef __attribute__((ext_vector_type(2))) float v2f;
typedef __attribute__((ext_vector_type(8))) float v8f;

// A (16x4, MxK) fragment: lane l -> m = l&15 ; holds K = {2*(l>>4), 2*(l>>4)+1}
__device__ __forceinline__ v2f ld_a(const float* A, int lda, int row0, int k, int lane) {
  const float* p = A + (row0 + (lane & 15)) * lda + k + ((lane >> 4) << 1);
  return v2f{p[0], p[1]};
}

// Accumulate A(64xK) * B(K x *)[:, colblk:colblk+128] into this wave's 4 subtiles.
// B chunks are staged k-major in LDS: Bs[n][kk] = Bm[(kb+kk)*ldb + colblk + n].
//  - staging: coalesced global b32 x4 -> ds_store_b128
//  - fragment read: one ds_load_b64 per subtile per k-step, bank-conflict-free
__device__ __forceinline__ void gemm_staged(v8f acc[4], const float* __restrict__ A, int lda,
                                            const float* __restrict__ Bm, int ldb, int K,
                                            int row0, int colblk, int col0w, int lane, int tid) {
  __shared__ float Bs[128][PADB];
  const int n_st  = tid & 127;        // staging column within tile
  const int k_st0 = (tid >> 7) * 4;   // 0 or 4

  for (int kb = 0; kb < K; kb += KCH) {
    // ---- stage B chunk (64 x 128) into LDS, transposed to k-major ----
#pragma unroll
    for (int kk = k_st0; kk < KCH; kk += 8) {
      const float* gp = Bm + (size_t)(kb + kk) * ldb + colblk + n_st;
      float4 v = {gp[0], gp[(size_t)ldb], gp[2 * (size_t)ldb], gp[3 * (size_t)ldb]};
      *(float4*)&Bs[n_st][kk] = v;
    }
    // prefetch next chunk into cache while this one is consumed (global_prefetch_b8)
    if (kb + KCH < K)
      __builtin_prefetch(Bm + (size_t)(kb + KCH + k_st0) * ldb + colblk + n_st, 0, 1);
    __syncthreads();

    // ---- consume chunk: 16 k-steps x 4 subtiles of v_wmma_f32_16x16x4_f32 ----
#pragma unroll 4
    for (int k = 0; k < KCH; k += 4) {
      v2f a = ld_a(A, lda, row0, kb + k, lane);
#pragma unroll
      for (int i = 0; i < 4; ++i) {
        int n = col0w + 16 * i + (lane & 15);
        const float* p = &Bs[n][k + ((lane >> 4) << 1)];
        v2f b = {p[0], p[1]};
        acc[i] = __builtin_amdgcn_wmma_f32_16x16x4_f32(false, a, false, b, (short)0, acc[i],
                                                       false, false);
      }
    }
    __syncthreads();
  }
}

// init = x0 @ W_init + b_init ; blocks 0..15 -> s0 columns, blocks 16..31 -> h columns
__global__ void RSLP_init_kernel(const float* __restrict__ x0, const float* __restrict__ Wi,
                                 const float* __restrict__ bi, float* __restrict__ s0,
                                 float* __restrict__ h) {
  int tid = threadIdx.x, lane = tid & 31, wave = tid >> 5;
  int row0   = (wave & 3) * 16;
  int col0w  = (wave >> 2) * 64;
  int colblk = blockIdx.x * 128;
  v8f acc[4] = {};
  gemm_staged(acc, x0, DINIT, Wi, 2 * NN, DINIT, row0, colblk, col0w, lane, tid);

  float* dst = (blockIdx.x < 16) ? s0 : h;          // block-uniform split (no divergence)
  int joff   = (blockIdx.x < 16) ? 0 : NN;
#pragma unroll
  for (int i = 0; i < 4; ++i) {
    int j = colblk + col0w + 16 * i + (lane & 15);
    float bj = bi[j];
#pragma unroll
    for (int v = 0; v < 8; ++v) {
      int m = row0 + v + 8 * (lane >> 4);
      dst[m * NN + (j - joff)] = acc[i][v] + bj;
    }
  }
}

// One scan step (+ deferred output GEMM of the previous step).
// Blocks 0..15 : h/s update for 128 N-cols each (skipped when t==TT)
// Blocks 16..19: out_{t-1} = s_t @ o, 128 DOUT-cols each (skipped when t==0)
__global__ void RSLP_step_kernel(const float* __restrict__ x, const float* __restrict__ Win,
                                 const float* __restrict__ bin, const float* __restrict__ r,
                                 const float* __restrict__ rb, const float* __restrict__ o,
                                 const float* __restrict__ s_old, float* __restrict__ s_new,
                                 float* __restrict__ h, float* __restrict__ out, int t) {
  int tid = threadIdx.x, lane = tid & 31, wave = tid >> 5;
  int row0  = (wave & 3) * 16;
  int col0w = (wave >> 2) * 64;
  int bid = blockIdx.x;

  if (bid < 16) {
    if (t >= TT) return;                         // block-uniform: EXEC stays full
    int colblk = bid * 128;
    v8f acc[4] = {};
    // s_t @ r (full r; diagonal removed in epilogue)
    gemm_staged(acc, s_old, NN, r, NN, NN, row0, colblk, col0w, lane, tid);
    // + x[:,t,:] @ W_in  (row m of A is x[(m*TT + t)*DIN + k])
    gemm_staged(acc, x + t * DIN, TT * DIN, Win, NN, DIN, row0, colblk, col0w, lane, tid);
#pragma unroll
    for (int i = 0; i < 4; ++i) {
      int j = colblk + col0w + 16 * i + (lane & 15);
      float addj  = rb[j] + bin[j];
      float rdiag = r[j * NN + j];               // r_eff = r with zero diagonal
#pragma unroll
      for (int v = 0; v < 8; ++v) {
        int m = row0 + v + 8 * (lane >> 4);
        float pre = acc[i][v] - s_old[m * NN + j] * rdiag + addj;
        float hn  = (1.0f - DECAY) * h[m * NN + j] + DECAY * pre;
        h[m * NN + j]     = hn;
        s_new[m * NN + j] = fmaxf(hn, 0.0f);
      }
    }
  } else {
    if (t < 1) return;                           // block-uniform
    int colblk = (bid - 16) * 128;
    v8f acc[4] = {};
    gemm_staged(acc, s_old, NN, o, DOUT, NN, row0, colblk, col0w, lane, tid);  // s_t @ o
#pragma unroll
    for (int i = 0; i < 4; ++i) {
      int j = colblk + col0w + 16 * i + (lane & 15);
#pragma unroll
      for (int v = 0; v < 8; ++v) {
        int m = row0 + v + 8 * (lane >> 4);
        out[(m * TT + (t - 1)) * DOUT + j] = acc[i][v];
      }
    }
  }
}

extern "C" void kernel_launch(void* const* d_in, const int* in_sizes, int n_in,
                              void* d_out, int out_size, void* d_ws, size_t ws_size,
                              hipStream_t stream) {
  (void)in_sizes; (void)n_in; (void)out_size; (void)ws_size;
  const float* x      = (const float*)d_in[0];
  const float* x0     = (const float*)d_in[1];
  const float* W_init = (const float*)d_in[2];
  const float* b_init = (const float*)d_in[3];
  const float* W_in   = (const float*)d_in[4];
  const float* b_in   = (const float*)d_in[5];
  const float* r      = (const float*)d_in[6];
  const float* r_b    = (const float*)d_in[7];
  const float* o      = (const float*)d_in[8];
  float* out = (float*)d_out;

  // workspace: s ping-pong (2 x B*N) + h (B*N) = 1.5 MB
  float* sb0  = (float*)d_ws;
  float* sb1  = sb0 + BB * NN;
  float* hbuf = sb1 + BB * NN;
  float* sb[2] = {sb0, sb1};

  RSLP_init_kernel<<<32, 256, 0, stream>>>(x0, W_init, b_init, sb0, hbuf);
  // t = 0..127: update (s,h); emit out_{t-1}.  t = 128: emit out_{127} only.
  for (int t = 0; t <= TT; ++t) {
    RSLP_step_kernel<<<20, 256, 0, stream>>>(x, W_in, b_in, r, r_b, o,
                                             sb[t & 1], sb[(t + 1) & 1], hbuf, out, t);
  }
}